// CAttention_11647951307135
// MI455X (gfx1250) — compile-verified
//
#include <hip/hip_runtime.h>

typedef __attribute__((ext_vector_type(2))) float v2f;
typedef __attribute__((ext_vector_type(4))) float v4f;
typedef __attribute__((ext_vector_type(8))) float v8f;

#define BB 64
#define CC 8
#define NN 2048
#define TT 64
#define NT (NN * TT) // 131072 columns per (b, c) row

// ---------------------------------------------------------------------------
// Pass 1: k[b,c,t] = sum_i alpha[i] * x[b,c,i,t]
// One block per (b,c); 256 threads. Thread (i0, tq) owns t-quad tq (4 t's via
// float4 loads) and strides i by 16: every lane issues global_load_b128
// (512 B per wave per instruction), fully dense. 128 iterations/thread.
// Reads 256 MB total (~11 us at 23.3 TB/s).
// ---------------------------------------------------------------------------
__global__ __launch_bounds__(256) void kReduceAlpha(
    const float* __restrict__ x, const float* __restrict__ alpha,
    float* __restrict__ kout) {
  __shared__ v4f red[16][16]; // [i0][tq]
  const int tid = threadIdx.x;
  const int tq  = tid & 15;  // t-quad: t = 4*tq .. 4*tq+3
  const int i0  = tid >> 4;  // 0..15
  const int bc  = blockIdx.x;
  const float* xp = x + (size_t)bc * NN * TT + (size_t)tq * 4;
  v4f acc = {0.f, 0.f, 0.f, 0.f};
  for (int i = i0; i < NN; i += 16) {
    const float a  = alpha[i];
    const v4f  xv = *(const v4f*)(xp + (size_t)i * TT);
    acc.x = fmaf(a, xv.x, acc.x);
    acc.y = fmaf(a, xv.y, acc.y);
    acc.z = fmaf(a, xv.z, acc.z);
    acc.w = fmaf(a, xv.w, acc.w);
  }
  red[i0][tq] = acc;
  __syncthreads();
  if (tid < 16) { // thread tq reduces its column of 16 partials
    v4f s = red[0][tid];
#pragma unroll
    for (int j = 1; j < 16; ++j) {
      const v4f r = red[j][tid];
      s.x += r.x; s.y += r.y; s.z += r.z; s.w += r.w;
    }
    *(v4f*)(kout + (size_t)bc * TT + (size_t)tid * 4) = s;
  }
}

// ---------------------------------------------------------------------------
// Pass 2: per batch b: tmp = k[b] @ Wc (8x64), scores = tmp @ k[b]^T (8x8),
// att[b] = softmax(scores, axis=-1). Tiny (~2 MFLOP total): one 64-thread
// block per batch, everything staged in LDS.
// ---------------------------------------------------------------------------
__global__ __launch_bounds__(64) void kScoresSoftmax(
    const float* __restrict__ kin, const float* __restrict__ Wc,
    float* __restrict__ att) {
  __shared__ float kl[CC][TT];
  __shared__ float tmp[CC][TT];
  __shared__ float sc[CC][CC];
  const int b = blockIdx.x;
  const int s = threadIdx.x; // 0..63
  for (int c = 0; c < CC; ++c)
    kl[c][s] = kin[((size_t)b * CC + c) * TT + s];
  __syncthreads();
  // tmp[c][s] = sum_t kl[c][t] * Wc[t][s]
  for (int c = 0; c < CC; ++c) {
    float acc = 0.f;
    for (int t = 0; t < TT; ++t)
      acc = fmaf(kl[c][t], Wc[t * TT + s], acc);
    tmp[c][s] = acc;
  }
  __syncthreads();
  { // scores[c][d] = sum_s tmp[c][s] * kl[d][s]; thread s -> (c,d)
    const int c = s >> 3, d = s & 7;
    float acc = 0.f;
    for (int q = 0; q < TT; ++q)
      acc = fmaf(tmp[c][q], kl[d][q], acc);
    sc[c][d] = acc;
  }
  __syncthreads();
  if (s < CC) { // row softmax, one thread per row
    const int c = s;
    float m = sc[c][0];
    for (int d = 1; d < CC; ++d) m = fmaxf(m, sc[c][d]);
    float e[CC], sum = 0.f;
    for (int d = 0; d < CC; ++d) { e[d] = __expf(sc[c][d] - m); sum += e[d]; }
    const float inv = 1.f / sum;
    for (int d = 0; d < CC; ++d)
      att[((size_t)b * CC + c) * CC + d] = e[d] * inv;
  }
}

// ---------------------------------------------------------------------------
// Pass 3: out[b,c,col] = sum_i att[b,c,i] * x[b,i,col]   (col = n*T+t flat)
// WMMA mapping, two batches packed per 16x16 tile:
//   A (16x16, via 4 chunks of V_WMMA_F32_16X16X4_F32) = diag(att[b0], att[b1])
//   B rows 0..7  = x[b0] channel rows, rows 8..15 = x[b1] channel rows
//     (contiguous in memory since x is [B,C,N,T])
//   D rows 0..7 -> out[b0], rows 8..15 -> out[b1]
// Layouts per ISA 7.12.2: A/B VGPR0 = K (lanes 0-15) / K+2 (lanes 16-31),
// VGPR1 = K+1 / K+3; C/D VGPRv = row v (lanes 0-15) / row v+8 (lanes 16-31).
// Stores are non-temporal: out is never re-read, keep L2 for the x stream.
// ---------------------------------------------------------------------------
#define TILES_PER_WAVE 4

__global__ __launch_bounds__(256) void kAggregateWmma(
    const float* __restrict__ x, const float* __restrict__ att,
    float* __restrict__ out) {
  const int lane = threadIdx.x & 31;
  const int wave = threadIdx.x >> 5; // 0..7
  const int bp   = blockIdx.y;       // 0..31 -> batches (2bp, 2bp+1)
  const int b0   = bp * 2;
  const int m    = lane & 15;
  const int half = lane >> 4;

  // Build block-diagonal A fragments once per wave.
  v2f a[4];
#pragma unroll
  for (int w = 0; w < 4; ++w) {
    const int kA0 = 4 * w + 2 * half;
    const int kA1 = kA0 + 1;
    float a0 = 0.f, a1 = 0.f;
    if (m < 8) {
      if (kA0 < 8) a0 = att[((size_t)b0 * CC + m) * CC + kA0];
      if (kA1 < 8) a1 = att[((size_t)b0 * CC + m) * CC + kA1];
    } else {
      if (kA0 >= 8) a0 = att[((size_t)(b0 + 1) * CC + (m - 8)) * CC + (kA0 - 8)];
      if (kA1 >= 8) a1 = att[((size_t)(b0 + 1) * CC + (m - 8)) * CC + (kA1 - 8)];
    }
    a[w].x = a0;
    a[w].y = a1;
  }

  // Batch-pair bases: rows 0..15 of the packed view are contiguous C-rows.
  const float* xp = x + (size_t)b0 * CC * NT;
  float* op = out + (size_t)b0 * CC * NT;

  const int tile0 = ((int)blockIdx.x * 8 + wave) * TILES_PER_WAVE;
#pragma unroll 2
  for (int tt = 0; tt < TILES_PER_WAVE; ++tt) {
    const size_t col = (size_t)(tile0 + tt) * 16 + m;
    // Load B fragments: chunk w covers packed rows 4w..4w+3.
    v2f bf[4];
#pragma unroll
    for (int w = 0; w < 4; ++w) {
      const int r0 = 4 * w + 2 * half;
      bf[w].x = xp[(size_t)r0 * NT + col];
      bf[w].y = xp[(size_t)(r0 + 1) * NT + col];
    }
    v8f d = {0.f, 0.f, 0.f, 0.f, 0.f, 0.f, 0.f, 0.f};
#pragma unroll
    for (int w = 0; w < 4; ++w)
      d = __builtin_amdgcn_wmma_f32_16x16x4_f32(false, a[w], false, bf[w],
                                                (short)0, d, false, false);
    // Store D: VGPR v holds packed row v (lanes 0-15) and v+8 (lanes 16-31).
#pragma unroll
    for (int v = 0; v < 8; ++v)
      __builtin_nontemporal_store(d[v], op + (size_t)(v + 8 * half) * NT + col);
  }
}

// ---------------------------------------------------------------------------
extern "C" void kernel_launch(void* const* d_in, const int* in_sizes, int n_in,
                              void* d_out, int out_size, void* d_ws,
                              size_t ws_size, hipStream_t stream) {
  const float* x     = (const float*)d_in[0];
  const float* Wc    = (const float*)d_in[1];
  const float* alpha = (const float*)d_in[2];
  float* out = (float*)d_out;

  float* kws   = (float*)d_ws;          // B*C*T  = 32768 floats (128 KB)
  float* attws = kws + BB * CC * TT;    // B*C*C  =  4096 floats ( 16 KB)

  kReduceAlpha<<<dim3(BB * CC), dim3(256), 0, stream>>>(x, alpha, kws);
  kScoresSoftmax<<<dim3(BB), dim3(64), 0, stream>>>(kws, Wc, attws);
  // tiles per batch-pair = NT/16 = 8192; per block = 8 waves * 4 = 32
  kAggregateWmma<<<dim3(NT / 16 / (8 * TILES_PER_WAVE), BB / 2), dim3(256), 0,
                   stream>>>(x, attws, out);
}